// SAGE_7739531067740
// MI455X (gfx1250) — compile-verified
//
#include <hip/hip_runtime.h>

// GCN 3-layer pipeline for MI455X (gfx1250, wave32).
// Working set (node features 25.6MB, agg 25.6MB, W 16KB) is L2-resident
// (192MB L2), so the edge gather (256MB/layer) and f32 atomic scatter
// (256MB/layer) run at L2 speed, not HBM. GEMM (~2.5 GFLOP total) is done
// with fp32 WMMA (v_wmma_f32_16x16x4_f32) to match the fp32 reference.

#define NNODES 100000
#define NEDGES 1000000
#define DF 64

typedef __attribute__((ext_vector_type(2))) float v2f;
typedef __attribute__((ext_vector_type(8))) float v8f;

__global__ void zero_f4_kernel(float4* __restrict__ p, int n4) {
  int i = blockIdx.x * blockDim.x + threadIdx.x;
  if (i < n4) {
    float4 z; z.x = 0.f; z.y = 0.f; z.z = 0.f; z.w = 0.f;
    p[i] = z;
  }
}

__global__ void deg_count_kernel(const int* __restrict__ src, const int* __restrict__ dst,
                                 float* __restrict__ odeg, float* __restrict__ ideg, int ne) {
  int e = blockIdx.x * blockDim.x + threadIdx.x;
  if (e < ne) {
    // f32 counting is exact below 2^24; max degree <= 1e6.
    __hip_atomic_fetch_add(&odeg[src[e]], 1.0f, __ATOMIC_RELAXED, __HIP_MEMORY_SCOPE_AGENT);
    __hip_atomic_fetch_add(&ideg[dst[e]], 1.0f, __ATOMIC_RELAXED, __HIP_MEMORY_SCOPE_AGENT);
  }
}

__global__ void deg_to_norm_kernel(float* __restrict__ buf, int n) {
  int i = blockIdx.x * blockDim.x + threadIdx.x;
  if (i < n) buf[i] = rsqrtf(fmaxf(buf[i], 1.0f));
}

// P[i,:] = X[i,:] * norm[i], vectorized as float4 (16 float4 per 64-wide row).
__global__ void scale_rows_kernel(const float4* __restrict__ X, const float* __restrict__ norm,
                                  float4* __restrict__ P, int n4) {
  int i = blockIdx.x * blockDim.x + threadIdx.x;
  if (i < n4) {
    float s = norm[i >> 4];
    float4 v = X[i];
    v.x *= s; v.y *= s; v.z *= s; v.w *= s;
    P[i] = v;
  }
}

// One wave32 per edge; each lane moves a float2 (64 floats/edge) with native
// global_atomic_add_f32 (relaxed, agent scope -> no CAS loop).
__global__ void scatter_add_kernel(const int* __restrict__ src, const int* __restrict__ dst,
                                   const float* __restrict__ P, float* __restrict__ Q, int ne) {
  int e = (blockIdx.x * blockDim.x + threadIdx.x) >> 5;
  int lane = threadIdx.x & 31;
  if (e < ne) {
    int s = src[e];
    int d = dst[e];
    float2 h = ((const float2*)(P + (size_t)s * DF))[lane];
    float* q = Q + (size_t)d * DF + lane * 2;
    __hip_atomic_fetch_add(q,     h.x, __ATOMIC_RELAXED, __HIP_MEMORY_SCOPE_AGENT);
    __hip_atomic_fetch_add(q + 1, h.y, __ATOMIC_RELAXED, __HIP_MEMORY_SCOPE_AGENT);
  }
}

// Y[rows 16w..16w+15, :] = act( (A[rows,:] * innorm[rows]) @ W + bias )
// One wave owns a full 16x64 output block (4 column tiles of 16x16), so the
// in-place update over the aggregation buffer is race-free: A reads feed the
// WMMA accumulator chain before any store, and no other wave touches the block.
//
// V_WMMA_F32_16X16X4_F32 fragment layout (ISA 7.12.2):
//   A 16x4: lane L holds row (L&15), K = kb + 2*(L>>4) + {0,1}  -> float2 load
//   B 4x16: lane L holds col (L&15), K = kb + 2*(L>>4) + {0,1}
//   C/D   : VGPR r <-> row 8*(L>>4)+r, col (L&15)
__global__ void gemm_wmma_kernel(const float* A, const float* __restrict__ innorm,
                                 const float* __restrict__ W, const float* __restrict__ bias,
                                 float* Y, int ntiles, int do_relu) {
  int wave = (blockIdx.x * blockDim.x + threadIdx.x) >> 5;
  int lane = threadIdx.x & 31;
  if (wave >= ntiles) return;            // wave-uniform: active waves keep EXEC all-ones
  int l  = lane & 15;
  int hi = lane >> 4;
  int row = wave * 16 + l;
  float s = innorm[row];
  const float* arow  = A + (size_t)row * DF + 2 * hi;   // K offset for this half-wave
  const float* wbase = W + 2 * hi * DF + l;

  v8f c0 = {}, c1 = {}, c2 = {}, c3 = {};
#pragma unroll
  for (int kb = 0; kb < DF; kb += 4) {
    float2 av = *(const float2*)(arow + kb);            // A[row][kb+2hi], [kb+2hi+1]
    v2f a; a[0] = av.x * s; a[1] = av.y * s;            // fold in_norm into A
    const float* wk = wbase + kb * DF;
    v2f b;
    b[0] = wk[0];        b[1] = wk[DF];
    c0 = __builtin_amdgcn_wmma_f32_16x16x4_f32(false, a, false, b, (short)0, c0, false, false);
    b[0] = wk[16];       b[1] = wk[DF + 16];
    c1 = __builtin_amdgcn_wmma_f32_16x16x4_f32(false, a, false, b, (short)0, c1, false, false);
    b[0] = wk[32];       b[1] = wk[DF + 32];
    c2 = __builtin_amdgcn_wmma_f32_16x16x4_f32(false, a, false, b, (short)0, c2, false, false);
    b[0] = wk[48];       b[1] = wk[DF + 48];
    c3 = __builtin_amdgcn_wmma_f32_16x16x4_f32(false, a, false, b, (short)0, c3, false, false);
  }

  float bb0 = bias[l], bb1 = bias[16 + l], bb2 = bias[32 + l], bb3 = bias[48 + l];
#pragma unroll
  for (int r = 0; r < 8; ++r) {
    int orow = wave * 16 + hi * 8 + r;
    float* yr = Y + (size_t)orow * DF + l;
    float v0 = c0[r] + bb0;
    float v1 = c1[r] + bb1;
    float v2 = c2[r] + bb2;
    float v3 = c3[r] + bb3;
    if (do_relu) {
      v0 = fmaxf(v0, 0.f); v1 = fmaxf(v1, 0.f);
      v2 = fmaxf(v2, 0.f); v3 = fmaxf(v3, 0.f);
    }
    yr[0] = v0; yr[16] = v1; yr[32] = v2; yr[48] = v3;
  }
}

extern "C" void kernel_launch(void* const* d_in, const int* in_sizes, int n_in,
                              void* d_out, int out_size, void* d_ws, size_t ws_size,
                              hipStream_t stream) {
  const float* x   = (const float*)d_in[0];
  const int*   src = (const int*)  d_in[1];
  const int*   dst = (const int*)  d_in[2];
  const float* W1  = (const float*)d_in[3];
  const float* b1  = (const float*)d_in[4];
  const float* W2  = (const float*)d_in[5];
  const float* b2  = (const float*)d_in[6];
  const float* W3  = (const float*)d_in[7];
  const float* b3  = (const float*)d_in[8];
  float* out = (float*)d_out;

  // Workspace layout: [out_norm N][in_norm N][P = N*D staging] ~= 26.4 MB
  char* ws = (char*)d_ws;
  float* out_norm = (float*)ws;
  float* in_norm  = out_norm + NNODES;
  float* P        = (float*)(ws + (size_t)2 * NNODES * sizeof(float)); // 800000B, 16B aligned

  const int T = 256;
  const int nd4     = NNODES * DF / 4;   // 1,600,000 float4 in N x D
  const int deg4    = 2 * NNODES / 4;    // 50,000 float4 covering both degree arrays
  const int rowtiles = NNODES / 16;      // 6250

  // ---- degree norms (shared by all 3 layers) ----
  zero_f4_kernel<<<(deg4 + T - 1) / T, T, 0, stream>>>((float4*)out_norm, deg4);
  deg_count_kernel<<<(NEDGES + T - 1) / T, T, 0, stream>>>(src, dst, out_norm, in_norm, NEDGES);
  deg_to_norm_kernel<<<(2 * NNODES + T - 1) / T, T, 0, stream>>>(out_norm, 2 * NNODES);

  const float* Ws[3] = {W1, W2, W3};
  const float* bs[3] = {b1, b2, b3};
  const float* Xin = x;

  for (int layer = 0; layer < 3; ++layer) {
    // h = X * out_norm  -> P
    scale_rows_kernel<<<(nd4 + T - 1) / T, T, 0, stream>>>((const float4*)Xin, out_norm,
                                                           (float4*)P, nd4);
    // agg (= d_out) := 0
    zero_f4_kernel<<<(nd4 + T - 1) / T, T, 0, stream>>>((float4*)out, nd4);
    // agg[dst] += P[src]   (one wave per edge -> 32M lanes total)
    scatter_add_kernel<<<NEDGES * 32 / T, T, 0, stream>>>(src, dst, P, out, NEDGES);
    // out = act( (agg * in_norm) @ W + b )  in place (row-block exclusive per wave)
    gemm_wmma_kernel<<<(rowtiles * 32 + T - 1) / T, T, 0, stream>>>(
        out, in_norm, Ws[layer], bs[layer], out, rowtiles, layer < 2 ? 1 : 0);
    Xin = out;
  }
}